// SSIMLoss_15556371546119
// MI455X (gfx1250) — compile-verified
//
#include <hip/hip_runtime.h>
#include <math.h>

// ---------------------------------------------------------------------------
// SSIM loss, fully fused for MI455X (gfx1250, wave32).
//
// Design:
//  * single HBM pass over the two 64 MiB inputs (roofline floor ~5.5us);
//    all blur intermediates live in LDS / VGPRs.
//  * separable 11-tap Gaussian expressed as banded 16x16x28 fp32 matmuls on
//    the matrix pipe: V_WMMA_F32_16X16X4_F32, 7 k-steps per 16x16 tile.
//    fp32 WMMA (not bf16): SSIM variance terms uxx-ux^2 cancel against
//    c2=9e-4 and need fp32 accumulation AND fp32 inputs.
//  * per-block partial sums -> d_ws, deterministic tree reduction in a
//    finalize kernel (no float atomics => bitwise-stable across replays).
// ---------------------------------------------------------------------------

typedef __attribute__((ext_vector_type(2))) float v2f;
typedef __attribute__((ext_vector_type(8))) float v8f;

#define IMG   512
#define TR    32          // output rows per block
#define TC    64          // output cols per block
#define FROWS 48          // halo rows (42 valid, rest zero) rounded to 3x16
#define FSTR  76          // 74 valid cols + 2 zero pad (even => v2f aligned)
#define HROWS 48
#define HSTR  68          // 64 cols + pad

#define NBLOCKS ((IMG/TC)*(IMG/TR)*64)   // 8*16*64 = 8192

__device__ __forceinline__ float gaussw(int d, float invZ) {
  // normalized gaussian tap w[d] for d in [0,11), else 0 (sigma=1.5, K=11)
  if (d < 0 || d > 10) return 0.0f;
  float t = (float)(d - 5) * (1.0f / 1.5f);
  return __expf(-0.5f * t * t) * invZ;
}

__global__ __launch_bounds__(256)
void ssim_main(const float* __restrict__ xlog,
               const float* __restrict__ yin,
               float* __restrict__ partial)
{
  __shared__ float Sb[FROWS * FSTR];   // sigmoid(x) halo tile
  __shared__ float Yb[FROWS * FSTR];   // y halo tile
  __shared__ float Hb[HROWS * HSTR];   // horizontally blurred channel

  const int tid  = threadIdx.x;
  const int lane = tid & 31;
  const int wid  = tid >> 5;          // 8 waves
  const int half = lane >> 4;         // 0: lanes 0-15, 1: lanes 16-31
  const int l15  = lane & 15;

  const int r0 = blockIdx.y * TR;
  const int c0 = blockIdx.x * TC;
  const float* xp = xlog + (size_t)blockIdx.z * IMG * IMG;
  const float* yp = yin  + (size_t)blockIdx.z * IMG * IMG;

  // ---- cooperative halo load + sigmoid (zero pad == conv zero padding) ----
  for (int idx = tid; idx < FROWS * FSTR; idx += 256) {
    int lr = idx / FSTR;
    int lc = idx - lr * FSTR;
    int gr = r0 - 5 + lr;
    int gc = c0 - 5 + lc;
    float s = 0.0f, yv = 0.0f;
    if (lr < 42 && lc < 74 && gr >= 0 && gr < IMG && gc >= 0 && gc < IMG) {
      float xv = xp[gr * IMG + gc];
      s  = 1.0f / (1.0f + __expf(-xv));
      yv = yp[gr * IMG + gc];
    }
    Sb[idx] = s;
    Yb[idx] = yv;
  }

  // ---- banded gaussian weight fragments (shared by H-pass B and V-pass A):
  //      frag[k] lane layout matches fp32 16x16x4 operand striping:
  //      element j of lane -> W[4k + 2*half + j][lane&15] = w[(m) - (lane&15)]
  float Z = 0.0f;
#pragma unroll
  for (int i = 0; i < 11; i++) {
    float t = (float)(i - 5) * (1.0f / 1.5f);
    Z += __expf(-0.5f * t * t);
  }
  const float invZ = 1.0f / Z;
  v2f wfrag[7];
#pragma unroll
  for (int k = 0; k < 7; k++) {
    int m0 = 4 * k + 2 * half;
    wfrag[k].x = gaussw(m0 - l15,     invZ);
    wfrag[k].y = gaussw(m0 + 1 - l15, invZ);
  }

  v8f u[5];   // blurred ux, uy, uxx, uyy, uxy for this wave's 16x16 out-tile

#pragma unroll
  for (int c = 0; c < 5; c++) {
    __syncthreads();   // previous vertical pass done reading Hb / initial load

    // ---- horizontal pass: 12 chunks (3 row-blocks x 4 col-blocks) ----
#pragma unroll
    for (int qq = 0; qq < 2; qq++) {
      int q = wid + qq * 8;
      if (q < 12) {                 // wave-uniform branch: EXEC stays all-1s
        int rb = (q >> 2) * 16;
        int cb = (q & 3) * 16;
        v8f acc = {0.f,0.f,0.f,0.f,0.f,0.f,0.f,0.f};
#pragma unroll
        for (int k = 0; k < 7; k++) {
          int row = rb + l15;
          int col = cb + 4 * k + 2 * half;          // even => 8B aligned
          const v2f s2 = *(const v2f*)&Sb[row * FSTR + col];
          const v2f y2 = *(const v2f*)&Yb[row * FSTR + col];
          v2f a;
          if      (c == 0) a = s2;
          else if (c == 1) a = y2;
          else if (c == 2) a = s2 * s2;
          else if (c == 3) a = y2 * y2;
          else             a = s2 * y2;
          acc = __builtin_amdgcn_wmma_f32_16x16x4_f32(
                    false, a, false, wfrag[k], (short)0, acc, false, false);
        }
#pragma unroll
        for (int v = 0; v < 8; v++)
          Hb[(rb + v + 8 * half) * HSTR + cb + l15] = acc[v];
      }
    }

    __syncthreads();   // Hb complete

    // ---- vertical pass: one 16x16 output chunk per wave (2x4 grid) ----
    {
      int rb = (wid >> 2) * 16;
      int cb = (wid & 3) * 16;
      v8f acc = {0.f,0.f,0.f,0.f,0.f,0.f,0.f,0.f};
#pragma unroll
      for (int k = 0; k < 7; k++) {
        int row = rb + 4 * k + 2 * half;
        v2f b;
        b.x = Hb[row * HSTR + cb + l15];
        b.y = Hb[(row + 1) * HSTR + cb + l15];
        acc = __builtin_amdgcn_wmma_f32_16x16x4_f32(
                  false, wfrag[k], false, b, (short)0, acc, false, false);
      }
      u[c] = acc;
    }
  }

  // ---- SSIM per pixel (8 accumulator rows per thread) ----
  const float c1 = 1e-4f;   // 0.01^2
  const float c2 = 9e-4f;   // 0.03^2
  float psum = 0.0f;
#pragma unroll
  for (int v = 0; v < 8; v++) {
    float ux  = u[0][v], uy  = u[1][v];
    float uxx = u[2][v], uyy = u[3][v], uxy = u[4][v];
    float vx  = uxx - ux * ux;
    float vy  = uyy - uy * uy;
    float vxy = uxy - ux * uy;
    float num = (2.0f * ux * uy + c1) * (2.0f * vxy + c2);
    float den = (ux * ux + uy * uy + c1) * (vx + vy + c2);
    psum += num / (den + 1e-12f);
  }

  // ---- deterministic block reduction (reuse Hb) ----
  __syncthreads();
  Hb[tid] = psum;
  __syncthreads();
#pragma unroll
  for (int s = 128; s > 0; s >>= 1) {
    if (tid < s) Hb[tid] += Hb[tid + s];
    __syncthreads();
  }
  if (tid == 0) {
    int blin = (blockIdx.z * gridDim.y + blockIdx.y) * gridDim.x + blockIdx.x;
    partial[blin] = Hb[0];
  }
}

__global__ __launch_bounds__(256)
void ssim_finalize(const float* __restrict__ partial, int n,
                   float* __restrict__ out)
{
  __shared__ float red[256];
  float s = 0.0f;
  for (int i = threadIdx.x; i < n; i += 256) s += partial[i];
  red[threadIdx.x] = s;
  __syncthreads();
#pragma unroll
  for (int k = 128; k > 0; k >>= 1) {
    if (threadIdx.x < k) red[threadIdx.x] += red[threadIdx.x + k];
    __syncthreads();
  }
  if (threadIdx.x == 0)
    out[0] = 1.0f - red[0] * (1.0f / 16777216.0f);   // N = 64*512*512
}

extern "C" void kernel_launch(void* const* d_in, const int* in_sizes, int n_in,
                              void* d_out, int out_size, void* d_ws, size_t ws_size,
                              hipStream_t stream)
{
  const float* x = (const float*)d_in[0];   // x_logits, fp32 [64,1,512,512]
  const float* y = (const float*)d_in[1];   // y,        fp32 [64,1,512,512]
  float* ws = (float*)d_ws;                 // needs NBLOCKS*4 = 32 KB scratch

  dim3 grid(IMG / TC, IMG / TR, 64);        // 8 x 16 x 64 = 8192 blocks
  ssim_main<<<grid, 256, 0, stream>>>(x, y, ws);
  ssim_finalize<<<1, 256, 0, stream>>>(ws, NBLOCKS, (float*)d_out);
}